// GCN_net_3375844294689
// MI455X (gfx1250) — compile-verified
//
#include <hip/hip_runtime.h>
#include <hip/hip_bf16.h>

typedef __attribute__((ext_vector_type(2))) float v2f;
typedef __attribute__((ext_vector_type(8))) float v8f;

// ---------------------------------------------------------------------------
// Degree / normalization kernels
// ---------------------------------------------------------------------------
__global__ void k_deg_init(float* __restrict__ deg, int n) {
    int i = blockIdx.x * blockDim.x + threadIdx.x;
    if (i < n) deg[i] = 1.0f;                       // self-loop contributes 1
}

__global__ void k_deg_accum(const int* __restrict__ dst, float* __restrict__ deg, int e) {
    int i = blockIdx.x * blockDim.x + threadIdx.x;
    if (i < e) atomicAdd(&deg[dst[i]], 1.0f);
}

__global__ void k_dinv(const float* __restrict__ deg, float* __restrict__ dinv, int n) {
    int i = blockIdx.x * blockDim.x + threadIdx.x;
    if (i < n) dinv[i] = rsqrtf(deg[i]);
}

// ---------------------------------------------------------------------------
// fp32 WMMA GEMM:  H[nrows x DCOL] = X[nrows x 64] * W[64 x DCOL]
// 8 waves / block, one 16x16 output tile per wave, K-loop of 16 x V_WMMA_F32_16X16X4_F32.
// Layouts per CDNA5 ISA 7.12.2:
//   A (16x4 f32, 2 VGPRs): lane%16 = row M, lane/16 selects K-pair {0,1}/{2,3}
//   B (4x16 f32, 2 VGPRs): lane%16 = col N, lane/16 selects K-pair
//   C/D (8 VGPRs): vgpr r, lanes 0-15 -> (M=r, N=lane), lanes 16-31 -> (M=r+8, N=lane-16)
// ---------------------------------------------------------------------------
template <int DCOL>
__global__ __launch_bounds__(256) void k_gemm_wmma(const float* __restrict__ X,
                                                   const float* __restrict__ W,
                                                   float* __restrict__ H,
                                                   int nrows) {
    constexpr int COLT = DCOL / 16;   // wave tiles across columns (4 or 2)
    constexpr int ROWT = 8 / COLT;    // wave tiles down rows per block (2 or 4)
    const int wave = threadIdx.x >> 5;
    const int lane = threadIdx.x & 31;
    const int m0 = (blockIdx.x * ROWT + (wave / COLT)) * 16;
    const int n0 = (wave % COLT) * 16;
    if (m0 >= nrows) return;          // wave-uniform: EXEC stays all-ones

    const int lr = lane & 15;         // row (A) / col (B,C)
    const int lh = lane >> 4;         // half-wave select

    v8f c = {};
    const float* xrow = X + (long long)(m0 + lr) * 64;
#pragma unroll
    for (int k = 0; k < 64; k += 4) {
        v2f a, b;
        a.x = xrow[k + lh * 2 + 0];
        a.y = xrow[k + lh * 2 + 1];
        b.x = W[(k + lh * 2 + 0) * DCOL + n0 + lr];
        b.y = W[(k + lh * 2 + 1) * DCOL + n0 + lr];
        // 8 args: (neg_a, A, neg_b, B, c_mod, C, reuse_a, reuse_b)
        c = __builtin_amdgcn_wmma_f32_16x16x4_f32(false, a, false, b,
                                                  (short)0, c, false, false);
    }
#pragma unroll
    for (int r = 0; r < 8; ++r) {
        H[(long long)(m0 + r + lh * 8) * DCOL + n0 + lr] = c[r];
    }
}

// ---------------------------------------------------------------------------
// out[i,f] = H[i,f] * dinv[i]^2 + b[f]   (self-loop term + bias; accumulator init)
// ---------------------------------------------------------------------------
__global__ void k_self_bias(const float* __restrict__ H, const float* __restrict__ dinv,
                            const float* __restrict__ b, float* __restrict__ out,
                            int n, int D) {
    long long idx = (long long)blockIdx.x * blockDim.x + threadIdx.x;
    long long total = (long long)n * D;
    if (idx >= total) return;
    int i = (int)(idx / D);
    int f = (int)(idx % D);
    float di = dinv[i];
    out[idx] = H[idx] * di * di + b[f];
}

// ---------------------------------------------------------------------------
// Edge scatter: out[dst] += H[src] * dinv[src]*dinv[dst]
// 16-byte vector gather per thread, D/4 threads per edge (coalesced), fp32 atomics
// (all traffic fits in the 192 MB L2, so RMWs resolve at L2).
// ---------------------------------------------------------------------------
__global__ void k_scatter(const float* __restrict__ H, const int* __restrict__ src,
                          const int* __restrict__ dst, const float* __restrict__ dinv,
                          float* __restrict__ out, int e, int D) {
    const int tpe = D >> 2;                     // threads per edge
    long long gid = (long long)blockIdx.x * blockDim.x + threadIdx.x;
    long long total = (long long)e * tpe;
    if (gid >= total) return;
    int edge = (int)(gid / tpe);
    int f = (int)(gid % tpe) * 4;
    int s = src[edge], d = dst[edge];
    float norm = dinv[s] * dinv[d];
    const float4 hv = *(const float4*)(H + (long long)s * D + f);
    float* op = out + (long long)d * D + f;
    atomicAdd(op + 0, hv.x * norm);
    atomicAdd(op + 1, hv.y * norm);
    atomicAdd(op + 2, hv.z * norm);
    atomicAdd(op + 3, hv.w * norm);
}

__global__ void k_relu(float* __restrict__ a, long long total) {
    long long idx = (long long)blockIdx.x * blockDim.x + threadIdx.x;
    if (idx < total) a[idx] = fmaxf(a[idx], 0.0f);
}

// ---------------------------------------------------------------------------
// Launcher
// ---------------------------------------------------------------------------
extern "C" void kernel_launch(void* const* d_in, const int* in_sizes, int n_in,
                              void* d_out, int out_size, void* d_ws, size_t ws_size,
                              hipStream_t stream) {
    const int D_IN = 64, D_H = 64, D_OUT = 32;
    const int N = in_sizes[0] / D_IN;       // 50000
    const int E = in_sizes[1] / 2;          // 800000

    const float* x  = (const float*)d_in[0];
    const int*   ei = (const int*)d_in[1];  // [2,E] int32
    const float* W1 = (const float*)d_in[2];
    const float* b1 = (const float*)d_in[3];
    const float* W2 = (const float*)d_in[4];
    const float* b2 = (const float*)d_in[5];
    float* out = (float*)d_out;             // [N, 32] fp32

    const int* srcIdx = ei;
    const int* dstIdx = ei + E;

    // Workspace layout (floats): deg | dinv | h1 | out1 | h2   (~32.4 MB total)
    float* ws   = (float*)d_ws;
    float* deg  = ws;
    float* dinv = deg  + N;
    float* h1   = dinv + N;
    float* out1 = h1   + (long long)N * D_H;
    float* h2   = out1 + (long long)N * D_H;

    const int BT = 256;
    const int rowTiles = N / 16;            // 3125 (N divides evenly by 16)

    // 1) degrees + rsqrt
    k_deg_init<<<(N + BT - 1) / BT, BT, 0, stream>>>(deg, N);
    k_deg_accum<<<(E + BT - 1) / BT, BT, 0, stream>>>(dstIdx, deg, E);
    k_dinv<<<(N + BT - 1) / BT, BT, 0, stream>>>(deg, dinv, N);

    // 2) layer 1: h1 = x @ W1  (WMMA fp32)
    k_gemm_wmma<D_H><<<(rowTiles + 1) / 2, BT, 0, stream>>>(x, W1, h1, N);

    // 3) out1 = h1 * dinv^2 + b1, then scatter edges, then ReLU
    {
        long long tot = (long long)N * D_H;
        k_self_bias<<<(unsigned)((tot + BT - 1) / BT), BT, 0, stream>>>(h1, dinv, b1, out1, N, D_H);
        long long st = (long long)E * (D_H / 4);
        k_scatter<<<(unsigned)((st + BT - 1) / BT), BT, 0, stream>>>(h1, srcIdx, dstIdx, dinv, out1, E, D_H);
        k_relu<<<(unsigned)((tot + BT - 1) / BT), BT, 0, stream>>>(out1, tot);
    }

    // 4) layer 2: h2 = out1 @ W2  (WMMA fp32)
    k_gemm_wmma<D_OUT><<<(rowTiles + 3) / 4, BT, 0, stream>>>(out1, W2, h2, N);

    // 5) out = h2 * dinv^2 + b2, then scatter edges
    {
        long long tot = (long long)N * D_OUT;
        k_self_bias<<<(unsigned)((tot + BT - 1) / BT), BT, 0, stream>>>(h2, dinv, b2, out, N, D_OUT);
        long long st = (long long)E * (D_OUT / 4);
        k_scatter<<<(unsigned)((st + BT - 1) / BT), BT, 0, stream>>>(h2, srcIdx, dstIdx, dinv, out, E, D_OUT);
    }

    (void)n_in; (void)out_size; (void)ws_size;
}